// ExplicitDisplacementWithLearnedWeights_35948876267794
// MI455X (gfx1250) — compile-verified
//
#include <hip/hip_runtime.h>
#include <hip/hip_bf16.h>
#include <math.h>

#define EPS    1e-8f
#define LN_EPS 1e-5f
#define MAX_DISP 5.0f

#define B_  8
#define L_  1024
#define K_  32
#define D_  256
#define H_  8
#define DH_ 64
#define NSITE (B_ * L_)

typedef __attribute__((ext_vector_type(2))) float v2f;
typedef __attribute__((ext_vector_type(4))) float v4f;
typedef __attribute__((ext_vector_type(8))) float v8f;

__device__ __forceinline__ float wredsum(float v) {
    #pragma unroll
    for (int i = 16; i > 0; i >>= 1) v += __shfl_xor(v, i, 32);
    return v;
}
__device__ __forceinline__ float sigmoidf_(float x) { return 1.0f / (1.0f + __expf(-x)); }
__device__ __forceinline__ float gelu_exact(float x) {
    return 0.5f * x * (1.0f + erff(x * 0.7071067811865476f));
}

// ---------------------------------------------------------------------------
// Kernel 1: per-row (b,l) feature_std (ddof=1) over D=256 of latents.
// One wave per row; 8 waves per block; b128 loads.
// ---------------------------------------------------------------------------
__global__ void stats_kernel(const float* __restrict__ latents,
                             float* __restrict__ fstd) {
    const int wave = threadIdx.x >> 5;
    const int lane = threadIdx.x & 31;
    const int row  = blockIdx.x * 8 + wave;
    if (row >= NSITE) return;
    const v4f* p4 = (const v4f*)(latents + (size_t)row * D_);
    float s = 0.f, ss = 0.f;
    #pragma unroll
    for (int i = lane; i < D_ / 4; i += 32) {
        v4f v = p4[i];
        s  += v.x + v.y + v.z + v.w;
        ss += v.x * v.x + v.y * v.y + v.z * v.z + v.w * v.w;
    }
    s  = wredsum(s);
    ss = wredsum(ss);
    if (lane == 0) {
        float var = (ss - s * s / (float)D_) / (float)(D_ - 1);
        fstd[row] = sqrtf(fmaxf(var, 0.f));
    }
}

// ---------------------------------------------------------------------------
// Kernel 2: per-batch min/max of feature_std over L=1024.
// ---------------------------------------------------------------------------
__global__ void minmax_kernel(const float* __restrict__ fstd,
                              float* __restrict__ smin,
                              float* __restrict__ smax) {
    __shared__ float smn[256], smx[256];
    const int b = blockIdx.x;
    const int t = threadIdx.x;
    float mn = 3.402823e38f, mx = -3.402823e38f;
    for (int l = t; l < L_; l += 256) {
        float v = fstd[b * L_ + l];
        mn = fminf(mn, v); mx = fmaxf(mx, v);
    }
    smn[t] = mn; smx[t] = mx;
    __syncthreads();
    for (int o = 128; o > 0; o >>= 1) {
        if (t < o) {
            smn[t] = fminf(smn[t], smn[t + o]);
            smx[t] = fmaxf(smx[t], smx[t + o]);
        }
        __syncthreads();
    }
    if (t == 0) { smin[b] = smn[0]; smax[b] = smx[0]; }
}

// ---------------------------------------------------------------------------
// Kernel 3 (WMMA): h = GELU(latents @ w1^T + b1)   [M=8192, N=64, K=256], f32.
// V_WMMA_F32_16X16X4_F32; block = 128 thr = 4 waves, wave w owns N-slab.
// ---------------------------------------------------------------------------
__global__ void mlp_wmma_kernel(const float* __restrict__ latents,
                                const float* __restrict__ w1,   // [64,256]
                                const float* __restrict__ b1,   // [64]
                                float* __restrict__ h_out) {    // [8192,64]
    __shared__ __align__(16) float sA[16 * D_];
    const int row0 = blockIdx.x * 16;
    const int tid  = threadIdx.x;
    const int wave = tid >> 5;
    const int lane = tid & 31;
    const int r    = lane & 15;
    const int half = lane >> 4;
    const int n0   = wave * 16;

    {
        const v4f* src = (const v4f*)(latents + (size_t)row0 * D_);
        v4f* dst = (v4f*)sA;
        for (int i = tid; i < 16 * D_ / 4; i += 128) dst[i] = src[i];
    }
    __syncthreads();

    v8f acc = {};
    const float* wrow = w1 + (size_t)(n0 + r) * D_;
    #pragma unroll 4
    for (int kk = 0; kk < D_; kk += 4) {
        v2f a, bm;
        a.x  = sA[r * D_ + kk + 2 * half];
        a.y  = sA[r * D_ + kk + 2 * half + 1];
        bm.x = wrow[kk + 2 * half];
        bm.y = wrow[kk + 2 * half + 1];
        acc = __builtin_amdgcn_wmma_f32_16x16x4_f32(
            false, a, false, bm, (short)0, acc, false, false);
    }

    const int n = n0 + r;
    const float bias = b1[n];
    #pragma unroll
    for (int j = 0; j < 8; ++j) {
        int m = row0 + j + 8 * half;
        float v = acc[j] + bias;
        h_out[(size_t)m * DH_ + n] = gelu_exact(v);
    }
}

// ---------------------------------------------------------------------------
// Kernel 4: LayerNorm(h) then mods = h_ln @ w2^T + b2  (Dh=64 -> 3).
// ---------------------------------------------------------------------------
__global__ void lnmods_kernel(const float* __restrict__ h,     // [8192,64]
                              const float* __restrict__ ln_g,  // [64]
                              const float* __restrict__ ln_b,  // [64]
                              const float* __restrict__ w2,    // [3,64]
                              const float* __restrict__ b2,    // [3]
                              float* __restrict__ mods) {      // [8192,3]
    const int wave = threadIdx.x >> 5;
    const int lane = threadIdx.x & 31;
    const int row  = blockIdx.x * 8 + wave;
    if (row >= NSITE) return;
    const float* hr = h + (size_t)row * DH_;
    float h0 = hr[lane], h1 = hr[lane + 32];
    float mean = wredsum(h0 + h1) * (1.0f / DH_);
    float d0 = h0 - mean, d1 = h1 - mean;
    float var = wredsum(d0 * d0 + d1 * d1) * (1.0f / DH_);
    float rstd = rsqrtf(var + LN_EPS);
    float n0 = d0 * rstd * ln_g[lane]      + ln_b[lane];
    float n1 = d1 * rstd * ln_g[lane + 32] + ln_b[lane + 32];
    #pragma unroll
    for (int o = 0; o < 3; ++o) {
        float m = wredsum(n0 * w2[o * DH_ + lane] + n1 * w2[o * DH_ + lane + 32]);
        if (lane == 0) mods[row * 3 + o] = m + b2[o];
    }
}

// ---------------------------------------------------------------------------
// Kernel 5: main per-site kernel. One block (256 thr = 8 waves) per (b,l).
// Phase 1: each wave streams 4 k-rows of neighbor_features with NT b128
//          loads (512 B / wave / instruction); shuffle reductions.
// Phase 2: wave 0 (k = lane) does per-k math + final fuse.
// ---------------------------------------------------------------------------
__global__ void main_kernel(const float* __restrict__ latents,
                            const float* __restrict__ attn,     // [8,8,1024,33]
                            const float* __restrict__ nfeat,    // [8,1024,32,256]
                            const float* __restrict__ npos,     // [8,1024,32,2]
                            const float* __restrict__ cpos,     // [8,1024,2]
                            const float* __restrict__ head_w,   // [8]
                            const float* __restrict__ rep_temp,
                            const float* __restrict__ log_ba,
                            const float* __restrict__ log_br,
                            const float* __restrict__ imp_str,
                            const float* __restrict__ fstd,
                            const float* __restrict__ smin,
                            const float* __restrict__ smax,
                            const float* __restrict__ mods,     // [8192,3]
                            float* __restrict__ out) {          // [8,1024,2]
    __shared__ __align__(16) float s_lat[D_];
    __shared__ float s_dot[K_], s_nbsq[K_];

    const int site = blockIdx.x;
    const int b = site / L_;
    const int l = site - b * L_;
    const int tid  = threadIdx.x;
    const int wave = tid >> 5;
    const int lane = tid & 31;

    if (tid < D_ / 4)
        ((v4f*)s_lat)[tid] = ((const v4f*)(latents + (size_t)site * D_))[tid];
    __syncthreads();

    const v4f* lat4 = (const v4f*)s_lat;
    const float* nf = nfeat + (size_t)site * K_ * D_;
    #pragma unroll
    for (int kk = 0; kk < 4; ++kk) {
        const int k = wave * 4 + kk;
        const v4f* row4 = (const v4f*)(nf + (size_t)k * D_);
        float dot = 0.f, nb = 0.f;
        #pragma unroll
        for (int i = lane; i < D_ / 4; i += 32) {
            v4f f = __builtin_nontemporal_load(&row4[i]);   // single-use 256MB stream: NT
            v4f a = lat4[i];
            dot += f.x * a.x + f.y * a.y + f.z * a.z + f.w * a.w;
            nb  += f.x * f.x + f.y * f.y + f.z * f.z + f.w * f.w;
        }
        dot = wredsum(dot);
        nb  = wredsum(nb);
        if (lane == 0) { s_dot[k] = dot; s_nbsq[k] = nb; }
    }
    __syncthreads();

    if (wave != 0) return;
    const int k = lane;

    // ||latents||
    float ssq = 0.f;
    #pragma unroll
    for (int i = lane; i < D_ / 4; i += 32) {
        v4f a = lat4[i];
        ssq += a.x * a.x + a.y * a.y + a.z * a.z + a.w * a.w;
    }
    ssq = wredsum(ssq);
    const float na = fmaxf(sqrtf(ssq), EPS);

    const float nb  = fmaxf(sqrtf(s_nbsq[k]), EPS);
    const float sim = s_dot[k] / (na * nb);

    // geometry (b64 loads)
    const v2f cp = ((const v2f*)cpos)[site];
    const v2f np = ((const v2f*)npos)[(size_t)site * K_ + k];
    const float dx = np.x - cp.x, dy = np.y - cp.y;
    const float dist = sqrtf(dx * dx + dy * dy);

    // head-importance softmax (8 values, redundant per lane)
    float hw[H_];
    float mxw = -3.402823e38f;
    #pragma unroll
    for (int h = 0; h < H_; ++h) { hw[h] = head_w[h]; mxw = fmaxf(mxw, hw[h]); }
    float se = 0.f;
    #pragma unroll
    for (int h = 0; h < H_; ++h) { hw[h] = __expf(hw[h] - mxw); se += hw[h]; }
    const float inv_se = 1.0f / se;

    // weighted attention over heads, slice attn[:, :, :, 1:1+K]
    float wa = 0.f;
    #pragma unroll
    for (int h = 0; h < H_; ++h) {
        float aw = attn[(((size_t)b * H_ + h) * L_ + l) * (1 + K_) + 1 + k];
        wa += (hw[h] * inv_se) * aw;
    }
    float wa_sum = wredsum(wa);
    wa = wa / (wa_sum + EPS);

    // repulsion strength
    const float temp = fabsf(rep_temp[0]) + EPS;
    float rs = __expf(sim / temp) / (dist + 0.1f);
    float rs_sum = wredsum(rs);
    rs = rs / (rs_sum + EPS);
    const float inv_de = 1.0f / (dist + EPS);
    const float dax = -dx * inv_de, day = -dy * inv_de;

    // reductions over k
    const float wcx = wredsum(wa * np.x);
    const float wcy = wredsum(wa * np.y);
    const float rpx = wredsum(rs * dax);
    const float rpy = wredsum(rs * day);
    const float sim_sum = wredsum(sim);

    if (lane == 0) {
        const float attx = wcx - cp.x, atty = wcy - cp.y;
        const float uniq = 1.0f - sim_sum * (1.0f / K_);
        const float mn = smin[b], mx = smax[b];
        const float complexity = (fstd[site] - mn) / (mx - mn + EPS);
        const float importance = 0.5f * complexity + 0.5f * uniq;
        const float istr = sigmoidf_(imp_str[0]);
        const float explicit_scale = 1.0f - istr * importance;

        const float m0 = mods[site * 3 + 0];
        const float m1 = mods[site * 3 + 1];
        const float m2 = mods[site * 3 + 2];
        const float w_attn = __expf(log_ba[0]) * 2.0f * sigmoidf_(m0);
        const float w_rep  = __expf(log_br[0]) * 2.0f * sigmoidf_(m1);
        const float fscale = explicit_scale * sigmoidf_(m2);
        const float inv_tw = 1.0f / (w_attn + w_rep + EPS);

        v2f o;
        o.x = tanhf((w_attn * attx + w_rep * rpx) * inv_tw * fscale * (1.0f / MAX_DISP)) * MAX_DISP;
        o.y = tanhf((w_attn * atty + w_rep * rpy) * inv_tw * fscale * (1.0f / MAX_DISP)) * MAX_DISP;
        ((v2f*)out)[site] = o;
    }
}

// ---------------------------------------------------------------------------
extern "C" void kernel_launch(void* const* d_in, const int* in_sizes, int n_in,
                              void* d_out, int out_size, void* d_ws, size_t ws_size,
                              hipStream_t stream) {
    const float* latents  = (const float*)d_in[0];
    const float* attn     = (const float*)d_in[1];
    const float* nfeat    = (const float*)d_in[2];
    const float* npos     = (const float*)d_in[3];
    const float* cpos     = (const float*)d_in[4];
    const float* head_w   = (const float*)d_in[5];
    const float* rep_temp = (const float*)d_in[6];
    const float* log_ba   = (const float*)d_in[7];
    const float* log_br   = (const float*)d_in[8];
    const float* imp_str  = (const float*)d_in[9];
    const float* w1       = (const float*)d_in[10];
    const float* b1       = (const float*)d_in[11];
    const float* ln_g     = (const float*)d_in[12];
    const float* ln_b     = (const float*)d_in[13];
    const float* w2       = (const float*)d_in[14];
    const float* b2       = (const float*)d_in[15];
    float* out = (float*)d_out;

    // workspace layout (floats)
    float* ws      = (float*)d_ws;
    float* h_ws    = ws;                      // 8192 * 64
    float* mods_ws = h_ws + NSITE * DH_;      // 8192 * 3
    float* fstd_ws = mods_ws + NSITE * 3;     // 8192
    float* smin_ws = fstd_ws + NSITE;         // 8
    float* smax_ws = smin_ws + B_;            // 8

    stats_kernel<<<NSITE / 8, 256, 0, stream>>>(latents, fstd_ws);
    minmax_kernel<<<B_, 256, 0, stream>>>(fstd_ws, smin_ws, smax_ws);
    mlp_wmma_kernel<<<NSITE / 16, 128, 0, stream>>>(latents, w1, b1, h_ws);
    lnmods_kernel<<<NSITE / 8, 256, 0, stream>>>(h_ws, ln_g, ln_b, w2, b2, mods_ws);
    main_kernel<<<NSITE, 256, 0, stream>>>(latents, attn, nfeat, npos, cpos,
                                           head_w, rep_temp, log_ba, log_br, imp_str,
                                           fstd_ws, smin_ws, smax_ws, mods_ws, out);
}